// QueryAndGroup_3315714753182
// MI455X (gfx1250) — compile-verified
//
#include <hip/hip_runtime.h>

// Problem constants (from reference setup_inputs)
#define B_   8
#define N_   16384
#define M_   1024
#define C_   64
#define NS   32
#define R2   0.04f          // RADIUS^2 = 0.2^2
#define OUTC 67             // 3 xyz-diff channels + 64 feature channels

typedef float v2f __attribute__((ext_vector_type(2)));
typedef float v8f __attribute__((ext_vector_type(8)));

// ---------------------------------------------------------------------------
// Kernel A: ball query via V_WMMA_F32_16X16X4_F32.
// One wave (32 lanes) handles 16 consecutive query points.
// D[m][n] = |p_n|^2 - 2 q_m . p_n   computed as A(16x4) x B(4x16):
//   A[m] = (-2qx, -2qy, -2qz, 1),  B[.][n] = (px, py, pz, |p|^2)
// hit  <=>  D[m][n] < r^2 - |q_m|^2
// Two 16-point WMMA subtiles per iteration; the next iteration's 32 point
// coordinates are prefetched before the current tiles are processed, so the
// dependent global-load latency overlaps the WMMA + mask-extraction work.
// ---------------------------------------------------------------------------
__global__ __launch_bounds__(32) void ball_query_wmma_kernel(
    const float* __restrict__ xyz,      // (B, N, 3)
    const float* __restrict__ new_xyz,  // (B, M, 3)
    int* __restrict__ idx_out)          // (B, M, NS) in d_ws
{
    const int lane   = threadIdx.x;          // 0..31
    const int b      = blockIdx.x >> 6;      // M_/16 == 64 query groups per batch
    const int m_base = (blockIdx.x & 63) << 4;

    // ---- Build A matrix (queries), per ISA 16x4 f32 A layout ----
    const int mq = lane & 15;
    const float* qp = new_xyz + ((size_t)b * M_ + m_base + mq) * 3;
    const float qx = qp[0], qy = qp[1], qz = qp[2];
    const float thr = R2 - (qx * qx + qy * qy + qz * qz);

    v2f A;
    A.x = (lane < 16) ? (-2.0f * qx) : (-2.0f * qz);   // K=0 / K=2
    A.y = (lane < 16) ? (-2.0f * qy) : 1.0f;           // K=1 / K=3

    // Per-lane thresholds for accumulator VGPR v: query (v + 8*(lane>=16)).
    const int hbase = (lane >= 16) ? 8 : 0;
    float th[8];
#pragma unroll
    for (int v = 0; v < 8; ++v) th[v] = __shfl(thr, v + hbase, 32);

    int  cnt   = (lane < 16) ? 0 : NS;   // lanes 16..31 own no query
    int  first = N_ + 1;                 // reference pad value when no hits
    int* mylist = idx_out + ((size_t)b * M_ + m_base + mq) * NS;
    const float* pbase = xyz + (size_t)b * N_ * 3;
    const unsigned qsel  = (unsigned)(lane & 7);
    const int      shamt = (lane & 8) ? 16 : 0;
    const bool     owner = (lane < 16);

    // One WMMA subtile: compare + ballot + ascending-index extraction.
    auto subtile = [&](int tbase, float px, float py, float pz) {
        const float p2 = px * px + py * py + pz * pz;
        v2f Bm;
        Bm.x = owner ? px : pz;          // K=0 / K=2
        Bm.y = owner ? py : p2;          // K=1 / K=3
        v8f acc = {};
        acc = __builtin_amdgcn_wmma_f32_16x16x4_f32(
            false, A, false, Bm, (short)0, acc, false, false);

        unsigned mk[8];
#pragma unroll
        for (int v = 0; v < 8; ++v)
            mk[v] = (unsigned)__ballot(acc[v] < th[v]);

        // Lane m (<16) selects its query's 16-bit mask (uniform sources).
        unsigned t01 = (qsel & 1u) ? mk[1] : mk[0];
        unsigned t23 = (qsel & 1u) ? mk[3] : mk[2];
        unsigned t45 = (qsel & 1u) ? mk[5] : mk[4];
        unsigned t67 = (qsel & 1u) ? mk[7] : mk[6];
        unsigned t03 = (qsel & 2u) ? t23 : t01;
        unsigned t47 = (qsel & 2u) ? t67 : t45;
        unsigned sel = (qsel & 4u) ? t47 : t03;
        unsigned mask16 = owner ? ((sel >> shamt) & 0xFFFFu) : 0u;

        while (mask16 && cnt < NS) {
            const int bit = __builtin_ctz(mask16);
            mask16 &= mask16 - 1u;
            const int val = tbase + bit;
            if (cnt == 0) first = val;
            mylist[cnt] = val;
            ++cnt;
        }
    };

    // ---- Software-pipelined scan: 32 points/iter, prefetch one iter ahead.
    float cx0, cy0, cz0, cx1, cy1, cz1;
    {
        const float* p0 = pbase + (size_t)(0  + mq) * 3;
        const float* p1 = pbase + (size_t)(16 + mq) * 3;
        cx0 = p0[0]; cy0 = p0[1]; cz0 = p0[2];
        cx1 = p1[0]; cy1 = p1[1]; cz1 = p1[2];
    }
    for (int tile = 0; tile < N_; tile += 32) {
        // Prefetch next iteration's 32 points (issued before any wait on use).
        float nx0 = 0.f, ny0 = 0.f, nz0 = 0.f;
        float nx1 = 0.f, ny1 = 0.f, nz1 = 0.f;
        const int nt = tile + 32;
        if (nt < N_) {
            const float* p0 = pbase + (size_t)(nt + mq) * 3;
            const float* p1 = pbase + (size_t)(nt + 16 + mq) * 3;
            nx0 = p0[0]; ny0 = p0[1]; nz0 = p0[2];
            nx1 = p1[0]; ny1 = p1[1]; nz1 = p1[2];
        }

        subtile(tile,      cx0, cy0, cz0);
        subtile(tile + 16, cx1, cy1, cz1);

        // Uniform early exit once all 16 queries have NS hits.
        if ((unsigned)__ballot(cnt >= NS) == 0xFFFFFFFFu) break;

        cx0 = nx0; cy0 = ny0; cz0 = nz0;
        cx1 = nx1; cy1 = ny1; cz1 = nz1;
    }

    // Pad remaining slots with the first hit (or N+1 if no hits -> clamped
    // at gather time, mirroring JAX clip semantics).
    if (owner) {
        for (int k = cnt; k < NS; ++k) mylist[k] = first;
    }
}

// ---------------------------------------------------------------------------
// Kernel B: grouping. One block per (b, m). lane (t&31) = sample s, so each
// wave stores one contiguous 128B output row per channel iteration.
// out[b][c][m][s]: c<3 -> xyz[b][n][c] - new_xyz[b][m][c]; else features.
// ---------------------------------------------------------------------------
__global__ __launch_bounds__(256) void group_points_kernel(
    const float* __restrict__ xyz,      // (B, N, 3)
    const float* __restrict__ new_xyz,  // (B, M, 3)
    const float* __restrict__ feat,     // (B, C, N)
    const int*   __restrict__ idx,      // (B, M, NS)
    float* __restrict__ out)            // (B, 67, M, NS)
{
    const int bm = blockIdx.x;           // 0 .. B*M-1
    const int b  = bm >> 10;             // M_ == 1024
    const int m  = bm & 1023;
    const int s  = threadIdx.x & 31;
    const int c0 = threadIdx.x >> 5;     // 0..7

    int n = idx[(size_t)bm * NS + s];
    if (n > N_ - 1) n = N_ - 1;          // JAX clip for the no-neighbor case
    if (n < 0)      n = 0;

    const float* p  = xyz + ((size_t)b * N_ + n) * 3;
    const float* q  = new_xyz + (size_t)bm * 3;
    const float* fb = feat + (size_t)b * C_ * N_;
    const size_t obase = (size_t)b * OUTC * M_ * NS + (size_t)m * NS + s;

    for (int c = c0; c < OUTC; c += 8) {
        float v;
        if (c < 3) v = p[c] - q[c];
        else       v = fb[(size_t)(c - 3) * N_ + n];
        __builtin_nontemporal_store(v, out + obase + (size_t)c * ((size_t)M_ * NS));
    }
}

// ---------------------------------------------------------------------------
extern "C" void kernel_launch(void* const* d_in, const int* in_sizes, int n_in,
                              void* d_out, int out_size, void* d_ws, size_t ws_size,
                              hipStream_t stream) {
    const float* xyz     = (const float*)d_in[0];  // (8,16384,3)
    const float* new_xyz = (const float*)d_in[1];  // (8,1024,3)
    const float* feat    = (const float*)d_in[2];  // (8,64,16384)
    int*   idx = (int*)d_ws;                       // B*M*NS ints = 1 MB scratch
    float* out = (float*)d_out;                    // (8,67,1024,32)

    ball_query_wmma_kernel<<<dim3(B_ * (M_ / 16)), dim3(32), 0, stream>>>(
        xyz, new_xyz, idx);
    group_points_kernel<<<dim3(B_ * M_), dim3(256), 0, stream>>>(
        xyz, new_xyz, feat, idx, out);
}